// MSDeformAttn_18751827214870
// MI455X (gfx1250) — compile-verified
//
#include <hip/hip_runtime.h>
#include <hip/hip_bf16.h>

typedef __attribute__((ext_vector_type(2))) float v2f;
typedef __attribute__((ext_vector_type(8))) float v8f;

#define B__ 4
#define S__ 19950
#define C__ 256
#define NH__ 8
#define M__ (B__ * S__)   // 79800 rows
#define KK  256           // K for every projection
#define MB  128           // rows per block (8 waves x 16)
#define WPAD 260          // Wt row stride in floats -> conflict-free B reads
#define APAD 36           // At row stride in floats -> conflict-free A reads
#define KC  32            // K-chunk staged for A

#if defined(__gfx1250__) && __has_builtin(__builtin_amdgcn_global_load_async_to_lds_b128) && __has_builtin(__builtin_amdgcn_s_wait_asynccnt)
#define USE_ASYNC_LDS 1
typedef __attribute__((ext_vector_type(4))) int v4i_;
typedef __attribute__((address_space(3))) v4i_ lv4i_;
#else
#define USE_ASYNC_LDS 0
#endif

// ---------------------------------------------------------------------------
// LDS-staged fp32 WMMA GEMM:  Out[M,N] = A[M,K] @ Wm[N,K]^T + bias[N], K=256.
// Block: 8 waves -> 128 rows x (16*NT) cols. Weights live in LDS for the whole
// block; A is staged in coalesced K-chunks. Inner loop: ds_load_b64 fragments
// feeding NT independent V_WMMA_F32_16X16X4_F32 accumulator chains.
//
// fp32 WMMA fragment layout (16x16x4):
//   A 16x4 : lane<16 row=lane k0,k0+1 ; lane>=16 row=lane-16 k0+2,k0+3
//   B 4x16 : lane<16 col=lane k0,k0+1 ; lane>=16 col=lane-16 k0+2,k0+3
//   C/D    : vgpr r -> (row r + 8*(lane>=16), col lane&15)
// ---------------------------------------------------------------------------
template <int NT>
__global__ __launch_bounds__(256) void wmma_gemm_lds(
    const float* __restrict__ A, const float* __restrict__ Wm,
    const float* __restrict__ bias, float* __restrict__ Out,
    int M, int N, int groupsN)
{
  constexpr int NB = 16 * NT;
  __shared__ float Wt[NB][WPAD];
  __shared__ float At[MB][APAD];

  const int t    = threadIdx.x;
  const int lane = t & 31;
  const int wave = t >> 5;
  const int mb   = blockIdx.x / groupsN;
  const int ng   = blockIdx.x - mb * groupsN;
  const int mblk = mb * MB;
  const int n0   = ng * NB;
  const int half = lane >> 4;
  const int l16  = lane & 15;

  // ---- stage the full weight tile (NB x 256) into LDS ----
#if USE_ASYNC_LDS
  for (int i = t; i < NB * (KK / 4); i += 256) {
    const int r = i >> 6, c4 = i & 63;
    __builtin_amdgcn_global_load_async_to_lds_b128(
        (v4i_*)(Wm + (size_t)(n0 + r) * KK + c4 * 4),
        (lv4i_*)&Wt[r][c4 * 4], 0, 0);
  }
#else
  for (int i = t; i < NB * (KK / 4); i += 256) {
    const int r = i >> 6, c4 = i & 63;
    const float4 val = *(const float4*)(Wm + (size_t)(n0 + r) * KK + c4 * 4);
    *(float4*)&Wt[r][c4 * 4] = val;
  }
#endif

  v8f acc[NT];
#pragma unroll
  for (int j = 0; j < NT; ++j) acc[j] = (v8f){0.f,0.f,0.f,0.f,0.f,0.f,0.f,0.f};

  const float* ar = &At[wave * 16 + l16][half << 1];
  const float* bp[NT];
#pragma unroll
  for (int j = 0; j < NT; ++j) bp[j] = &Wt[j * 16 + l16][half << 1];

  for (int kc = 0; kc < KK; kc += KC) {
    __syncthreads();   // previous chunk's reads done (also first-iter fence)
    // ---- stage A chunk (MB x KC), fully coalesced b128 per row ----
#if USE_ASYNC_LDS
    for (int i = t; i < MB * (KC / 4); i += 256) {
      const int r = i >> 3, c4 = i & 7;
      int grow = mblk + r; if (grow > M - 1) grow = M - 1;
      __builtin_amdgcn_global_load_async_to_lds_b128(
          (v4i_*)(A + (size_t)grow * KK + kc + c4 * 4),
          (lv4i_*)&At[r][c4 * 4], 0, 0);
    }
    __builtin_amdgcn_s_wait_asynccnt(0);   // own async ops done before barrier
#else
    for (int i = t; i < MB * (KC / 4); i += 256) {
      const int r = i >> 3, c4 = i & 7;
      int grow = mblk + r; if (grow > M - 1) grow = M - 1;
      const float4 val = *(const float4*)(A + (size_t)grow * KK + kc + c4 * 4);
      *(float4*)&At[r][c4 * 4] = val;
    }
#endif
    __syncthreads();   // LDS tile visible to all waves

#pragma unroll
    for (int kk = 0; kk < KC; kk += 4) {
      const v2f a = *(const v2f*)(ar + kk);
#pragma unroll
      for (int j = 0; j < NT; ++j) {
        const v2f b = *(const v2f*)(bp[j] + kc + kk);
        acc[j] = __builtin_amdgcn_wmma_f32_16x16x4_f32(
            false, a, false, b, (short)0, acc[j], false, false);
      }
    }
  }

  // ---- epilogue: bias add + predicated store ----
  const int m0 = mblk + wave * 16;
  const int rbase = m0 + (half << 3);
#pragma unroll
  for (int j = 0; j < NT; ++j) {
    const int ncol = n0 + j * 16 + l16;
    const float bv = bias[ncol];
#pragma unroll
    for (int r = 0; r < 8; ++r) {
      const int row = rbase + r;
      if (row < M) Out[(size_t)row * N + ncol] = acc[j][r] + bv;
    }
  }
}

// ---------------------------------------------------------------------------
// Sampling: one wave per (b, s, head); lane = channel d (Dh == 32 == wave32).
// softmax over 4 points per (head, level), then bilinear gather from v.
//   ix = ref_x * W + off_x - 0.5   (algebraic simplification of the reference)
// ---------------------------------------------------------------------------
__device__ __forceinline__ float msda_corner(const float* __restrict__ vb,
                                             int Wl, int Hl, int xc, int yc,
                                             float w)
{
  const bool valid = (xc >= 0) & (xc < Wl) & (yc >= 0) & (yc < Hl);
  const int xcc = min(max(xc, 0), Wl - 1);
  const int ycc = min(max(yc, 0), Hl - 1);
  const float val = vb[(size_t)(ycc * Wl + xcc) * C__];
  return valid ? val * w : 0.0f;
}

__global__ __launch_bounds__(256) void msda_sample(
    const float* __restrict__ v,       // [M, 256]  (channel = h*32 + d)
    const float* __restrict__ off,     // [M, 192]
    const float* __restrict__ lg,      // [M, 96]
    const float* __restrict__ refp,    // [M, 2]
    float* __restrict__ outp)          // [M, 256]
{
  const int lane = threadIdx.x & 31;
  const int wave = threadIdx.x >> 5;
  const int idx  = blockIdx.x * 8 + wave;        // = m * 8 + h
  if (idx >= M__ * NH__) return;

  const int h = idx & 7;
  const int m = idx >> 3;                        // b*S + s
  const int b = m / S__;

  const int Hs[3] = {100, 50, 25};
  const int Ws[3] = {152, 76, 38};
  const int St[3] = {0, 15200, 19000};

  const float rx = refp[(size_t)m * 2 + 0];
  const float ry = refp[(size_t)m * 2 + 1];
  const float* offm = off + (size_t)m * 192 + h * 24;   // [l][p][2]
  const float* lgm  = lg  + (size_t)m * 96  + h * 12;   // [l][p]

  float acc = 0.0f;

#pragma unroll
  for (int l = 0; l < 3; ++l) {
    const int Hl = Hs[l], Wl = Ws[l];
    float e[4];
    float mx = lgm[l * 4 + 0];
#pragma unroll
    for (int p = 1; p < 4; ++p) mx = fmaxf(mx, lgm[l * 4 + p]);
    float esum = 0.0f;
#pragma unroll
    for (int p = 0; p < 4; ++p) { e[p] = expf(lgm[l * 4 + p] - mx); esum += e[p]; }
    const float inv = 1.0f / esum;

    const float* vb = v + ((size_t)b * S__ + St[l]) * C__ + h * 32 + lane;

#pragma unroll
    for (int p = 0; p < 4; ++p) {
      const float ox = offm[(l * 4 + p) * 2 + 0];
      const float oy = offm[(l * 4 + p) * 2 + 1];
      const float ix = rx * (float)Wl + ox - 0.5f;
      const float iy = ry * (float)Hl + oy - 0.5f;
      const float x0f = floorf(ix), y0f = floorf(iy);
      const int   x0 = (int)x0f,    y0 = (int)y0f;
      const float wx1 = ix - x0f,   wy1 = iy - y0f;
      const float wx0 = 1.0f - wx1, wy0 = 1.0f - wy1;
      float sv = 0.0f;
      sv += msda_corner(vb, Wl, Hl, x0,     y0,     wx0 * wy0);
      sv += msda_corner(vb, Wl, Hl, x0 + 1, y0,     wx1 * wy0);
      sv += msda_corner(vb, Wl, Hl, x0,     y0 + 1, wx0 * wy1);
      sv += msda_corner(vb, Wl, Hl, x0 + 1, y0 + 1, wx1 * wy1);
      acc += (e[p] * inv) * sv;
    }
  }

  outp[(size_t)m * C__ + h * 32 + lane] = acc;
}

// ---------------------------------------------------------------------------
extern "C" void kernel_launch(void* const* d_in, const int* in_sizes, int n_in,
                              void* d_out, int out_size, void* d_ws, size_t ws_size,
                              hipStream_t stream) {
  const float* query  = (const float*)d_in[0];
  const float* value  = (const float*)d_in[1];
  // d_in[2] spatial_shapes / d_in[3] level_start_index: compile-time constants
  const float* refp   = (const float*)d_in[4];
  const float* W_off  = (const float*)d_in[5];
  const float* b_off  = (const float*)d_in[6];
  const float* W_attn = (const float*)d_in[7];
  const float* b_attn = (const float*)d_in[8];
  const float* W_val  = (const float*)d_in[9];
  const float* b_val  = (const float*)d_in[10];
  const float* W_out  = (const float*)d_in[11];
  const float* b_out  = (const float*)d_in[12];
  float* out = (float*)d_out;

  float* ws   = (float*)d_ws;
  float* v    = ws;                                   // M*256 floats
  float* off  = v   + (size_t)M__ * 256;              // M*192
  float* lg   = off + (size_t)M__ * 192;              // M*96
  float* outp = lg  + (size_t)M__ * 96;               // M*256

  const int M = M__;
  const int tilesM = (M + MB - 1) / MB;               // 624

  // Stage 1: three independent projections (stream-ordered)
  wmma_gemm_lds<4><<<tilesM * 4, 256, 0, stream>>>(value, W_val,  b_val,  v,   M, 256, 4);
  wmma_gemm_lds<4><<<tilesM * 3, 256, 0, stream>>>(query, W_off,  b_off,  off, M, 192, 3);
  wmma_gemm_lds<2><<<tilesM * 3, 256, 0, stream>>>(query, W_attn, b_attn, lg,  M,  96, 3);

  // Stage 2: softmax + bilinear gather (one wave per (b,s,h))
  msda_sample<<<(M__ * NH__ + 7) / 8, 256, 0, stream>>>(v, off, lg, refp, outp);

  // Stage 3: output projection
  wmma_gemm_lds<4><<<tilesM * 4, 256, 0, stream>>>(outp, W_out, b_out, out, M, 256, 4);
}